// GraphConvolution_66984309948597
// MI455X (gfx1250) — compile-verified
//
#include <hip/hip_runtime.h>
#include <math.h>

typedef float v2f __attribute__((ext_vector_type(2)));
typedef float v8f __attribute__((ext_vector_type(8)));

#define NNODES 50000
#define NEDGES 1600000
#define NF     128
#define NKER   4
#define MTILES (NNODES / 16)              // 3125
#define NTILES (NF / 16)                  // 8
#define TOTAL_WAVES (MTILES * NTILES * NKER)  // 100000

// ---------------------------------------------------------------------------
// Repack weight (IN_F, OUT_F, KER) -> Wt[k][i][o] contiguous (4 x 128 x 128)
// ---------------------------------------------------------------------------
__global__ void wt_transpose_kernel(const float* __restrict__ w,
                                    float* __restrict__ wt) {
    int t = blockIdx.x * blockDim.x + threadIdx.x;   // 65536 elements
    int k  = t >> 14;
    int io = t & 16383;
    int i  = io >> 7;
    int o  = io & 127;
    wt[t] = w[i * (NF * NKER) + o * NKER + k];
}

// ---------------------------------------------------------------------------
// support[k] = x @ Wt[k]   via V_WMMA_F32_16X16X4_F32, one wave per 16x16 tile
// ---------------------------------------------------------------------------
__global__ void gemm_wmma_kernel(const float* __restrict__ x,
                                 const float* __restrict__ wt,
                                 float* __restrict__ support) {
    int wave = blockIdx.x * (blockDim.x >> 5) + (threadIdx.x >> 5);
    int lane = threadIdx.x & 31;
    if (wave >= TOTAL_WAVES) return;   // uniform per wave; never trips with exact grid

    int k   = wave / (MTILES * NTILES);
    int rem = wave % (MTILES * NTILES);
    int mt  = rem / NTILES;
    int nt  = rem % NTILES;

    const float* Wk = wt + (size_t)k * NF * NF;

    int l16  = lane & 15;
    int koff = (lane >> 4) << 1;          // 0 for lanes 0-15, 2 for lanes 16-31
    int m    = mt * 16 + l16;             // A row for this lane
    int n    = nt * 16 + l16;             // B/C/D column for this lane

    const float* arow = x + (size_t)m * NF;

    v8f acc = {};
    #pragma unroll 4
    for (int kb = 0; kb < NF; kb += 4) {
        // A 16x4: lane<16 holds K=kb+0,kb+1 ; lane>=16 holds K=kb+2,kb+3 (b64 load)
        v2f a;
        a.x = arow[kb + koff + 0];
        a.y = arow[kb + koff + 1];
        // B 4x16: VGPR0 = row K=kb+koff, VGPR1 = row K=kb+koff+1, col n
        v2f b;
        b.x = Wk[(size_t)(kb + koff + 0) * NF + n];
        b.y = Wk[(size_t)(kb + koff + 1) * NF + n];
        acc = __builtin_amdgcn_wmma_f32_16x16x4_f32(
                  /*neg_a=*/false, a, /*neg_b=*/false, b,
                  /*c_mod=*/(short)0, acc, /*reuse_a=*/false, /*reuse_b=*/false);
    }

    // D: VGPR j = row mt*16 + j + (lane>=16 ? 8 : 0), col n
    int mrow = mt * 16 + ((lane >> 4) << 3);
    float* outp = support + ((size_t)k * NNODES + mrow) * NF + n;
    #pragma unroll
    for (int j = 0; j < 8; ++j)
        outp[(size_t)j * NF] = acc[j];
}

// ---------------------------------------------------------------------------
// out[n][f] = bias[f]   (d_out is poisoned; this also applies the +bias term)
// ---------------------------------------------------------------------------
__global__ void init_out_kernel(float* __restrict__ out,
                                const float* __restrict__ bias) {
    int t = blockIdx.x * blockDim.x + threadIdx.x;   // NNODES*NF
    out[t] = bias[t & (NF - 1)];
}

// ---------------------------------------------------------------------------
// Edge scatter: one wave per edge, each lane owns 4 output features.
// support is L2-resident (102.4 MB < 192 MB L2), so gathers & atomics hit L2.
// ---------------------------------------------------------------------------
__global__ void edge_kernel(const float* __restrict__ x,
                            const int* __restrict__ edge_idx,
                            const float* __restrict__ support,
                            const float* __restrict__ mu,
                            const float* __restrict__ sig,
                            float* __restrict__ out) {
    int gid  = blockIdx.x * blockDim.x + threadIdx.x;
    int e    = gid >> 5;
    int lane = gid & 31;
    if (e >= NEDGES) return;

    int row = edge_idx[e];
    int col = edge_idx[NEDGES + e];

    const float* xr = x + (size_t)row * NF;
    const float* xc = x + (size_t)col * NF;
    float dx = xr[0] - xc[0];
    float dy = xr[1] - xc[1];
    float dz = xr[2] - xc[2];

    float v[NKER];
    #pragma unroll
    for (int k = 0; k < NKER; ++k) {
        float d0 = dx - mu[0 * NKER + k];
        float d1 = dy - mu[1 * NKER + k];
        float d2 = dz - mu[2 * NKER + k];
        v[k] = __expf(-0.5f * sig[k] * (d0 * d0 + d1 * d1 + d2 * d2));
    }

    int f = lane << 2;                          // 4 features per lane
    float4 acc = make_float4(0.f, 0.f, 0.f, 0.f);
    #pragma unroll
    for (int k = 0; k < NKER; ++k) {
        const float4 s = *(const float4*)(support +
                          ((size_t)k * NNODES + col) * NF + f);
        acc.x += v[k] * s.x;
        acc.y += v[k] * s.y;
        acc.z += v[k] * s.z;
        acc.w += v[k] * s.w;
    }

    float* o = out + (size_t)row * NF + f;
    atomicAdd(o + 0, acc.x);
    atomicAdd(o + 1, acc.y);
    atomicAdd(o + 2, acc.z);
    atomicAdd(o + 3, acc.w);
}

// ---------------------------------------------------------------------------
extern "C" void kernel_launch(void* const* d_in, const int* in_sizes, int n_in,
                              void* d_out, int out_size, void* d_ws, size_t ws_size,
                              hipStream_t stream) {
    const float* x    = (const float*)d_in[0];   // (N, 128) f32
    const int*   eidx = (const int*)  d_in[1];   // (2, E)   int
    const float* w    = (const float*)d_in[2];   // (128, 128, 4) f32
    const float* bias = (const float*)d_in[3];   // (128,)
    const float* mu   = (const float*)d_in[4];   // (3, 4)
    const float* sig  = (const float*)d_in[5];   // (1, 4)
    float* out = (float*)d_out;

    float* support = (float*)d_ws;                               // 4*N*128 f32
    float* wt      = support + (size_t)NKER * NNODES * NF;       // 4*128*128 f32

    // 1) repack weights: 65536 elems
    wt_transpose_kernel<<<256, 256, 0, stream>>>(w, wt);

    // 2) WMMA GEMMs: 100000 waves, 8 waves/block
    gemm_wmma_kernel<<<TOTAL_WAVES / 8, 256, 0, stream>>>(x, wt, support);

    // 3) init out with bias: 6.4M elems
    init_out_kernel<<<(NNODES * NF) / 256, 256, 0, stream>>>(out, bias);

    // 4) edge gather/scatter: 32 lanes per edge
    edge_kernel<<<(NEDGES * 32) / 256, 256, 0, stream>>>(x, eidx, support, mu, sig, out);
}